// LevelAttention_loss_34823594836063
// MI455X (gfx1250) — compile-verified
//
#include <hip/hip_runtime.h>
#include <hip/hip_bf16.h>
#include <math.h>

typedef __attribute__((ext_vector_type(2))) float v2f;
typedef __attribute__((ext_vector_type(8))) float v8f;

#define NB    64
#define NBOX  100
#define NLVL  5
// per-batch block layout for the bce kernel: lvl0:64, lvl1:16, lvl2:4, lvl3:1, lvl4:1
#define BLOCKS_PER_BATCH 86

// ---------------------------------------------------------------------------
// Kernel 1: per-batch box scan -> rects/flags/anyValid, zero accumulators.
// ---------------------------------------------------------------------------
__global__ __launch_bounds__(64)
void la_setup_kernel(const float* __restrict__ bboxs,
                     const int* __restrict__ hp, const int* __restrict__ wp,
                     float* __restrict__ accum, int4* __restrict__ rect,
                     int* __restrict__ flags, int* __restrict__ anyv) {
  int b = threadIdx.x;
  if (b >= NB) return;
  const float hf = (float)hp[0];
  const float wf = (float)wp[0];
  const float* bx = bboxs + (size_t)b * NBOX * 4;

  #pragma unroll
  for (int l = 0; l < NLVL; ++l) accum[b * NLVL + l] = 0.0f;

  bool anyValid = false;
  for (int n = 0; n < NBOX; ++n) {
    float x1 = bx[n*4+0], y1 = bx[n*4+1], x2 = bx[n*4+2], y2 = bx[n*4+3];
    anyValid = anyValid || ((x1 <= wf) && (y1 <= hf) && (x2 <= wf) && (y2 <= hf));
  }
  anyv[b] = anyValid ? 1 : 0;

  for (int l = 0; l < NLVL; ++l) {
    // thresholds in double, rounded to f32 (matches JAX weak-typed python scalars)
    double s = (double)(1 << (l + 5));
    float minA = (float)(s * s * 0.5);
    float maxA = (float)((s * 1.58) * (s * 1.58) * 2.0);

    bool found = false;
    float b0 = bx[0], b1 = bx[1], b2 = bx[2], b3 = bx[3];  // argmax(all-false)==0
    for (int n = 0; n < NBOX; ++n) {
      float x1 = bx[n*4+0], y1 = bx[n*4+1], x2 = bx[n*4+2], y2 = bx[n*4+3];
      bool valid = (x1 <= wf) && (y1 <= hf) && (x2 <= wf) && (y2 <= hf);
      float area = (x2 - x1) * (y2 - y1);
      bool lm = valid && (area >= minA) && (area <= maxA);
      if (lm && !found) { found = true; b0 = x1; b1 = y1; b2 = x2; b3 = y2; }
    }
    int Wl = 512 >> l;
    int Hl = 512 >> l;
    float sw = (float)Wl / wf;
    float sh = (float)Hl / hf;
    int xi1 = max((int)truncf(b0 * sw), 0);
    int yi1 = max((int)truncf(b1 * sh), 0);
    int xi2 = min((int)truncf(b2 * sw) + 1, Wl);
    int yi2 = min((int)truncf(b3 * sh) + 1, Hl);
    rect[b * NLVL + l] = make_int4(xi1, yi1, xi2, yi2);
    flags[b * NLVL + l] = found ? 1 : 0;
  }
}

// ---------------------------------------------------------------------------
// Kernel 2: streaming BCE partial sums. One block handles a 4K-element tile of
// one (batch, level). WMMA f32 16x16x4 (A=partials, B=ones) collapses each
// wave's 32 partials; 8 waves combine in LDS; one float atomic per block.
// ---------------------------------------------------------------------------
__global__ __launch_bounds__(256)
void la_bce_kernel(const float* __restrict__ att0, const float* __restrict__ att1,
                   const float* __restrict__ att2, const float* __restrict__ att3,
                   const float* __restrict__ att4,
                   float* __restrict__ accum, const int4* __restrict__ rect,
                   const int* __restrict__ flags) {
  int bid = blockIdx.x;
  int b = bid / BLOCKS_PER_BATCH;
  int r = bid % BLOCKS_PER_BATCH;
  int lvl, tile;
  if (r < 64)      { lvl = 0; tile = r; }
  else if (r < 80) { lvl = 1; tile = r - 64; }
  else if (r < 84) { lvl = 2; tile = r - 80; }
  else if (r < 85) { lvl = 3; tile = 0; }
  else             { lvl = 4; tile = 0; }

  const float* base;
  switch (lvl) {
    case 0: base = att0; break;
    case 1: base = att1; break;
    case 2: base = att2; break;
    case 3: base = att3; break;
    default: base = att4; break;
  }
  int log2W = 9 - lvl;
  int Wl    = 1 << log2W;
  int elems = 1 << (2 * log2W);
  int epb   = (lvl == 4) ? 1024 : 4096;          // elements per block
  const float* map = base + (size_t)b * elems;

  int4 rc  = rect[b * NLVL + lvl];
  bool any = flags[b * NLVL + lvl] != 0;

  float part = 0.0f;
  int start = tile * epb;
  int end   = start + epb;
  for (int idx = start + (int)threadIdx.x * 4; idx < end; idx += 1024) {
    if (idx + 1024 < end) __builtin_prefetch(map + idx + 1024, 0, 0);
    float4 p4 = *(const float4*)(map + idx);
    int row = idx >> log2W;
    int col = idx & (Wl - 1);
    bool rok = any && (row >= rc.y) && (row < rc.w);
    float vals[4] = {p4.x, p4.y, p4.z, p4.w};
    #pragma unroll
    for (int j = 0; j < 4; ++j) {
      float p = vals[j];
      int c = col + j;
      bool gt = rok && (c >= rc.x) && (c < rc.z);
      float lp  = fmaxf(logf(p),     -100.0f);
      float l1p = fmaxf(log1pf(-p),  -100.0f);
      part -= gt ? lp : l1p;
    }
  }

  // ---- wave32 reduction via V_WMMA_F32_16X16X4_F32 ----
  // Lane L contributes A[m = L%16, k = (L<16 ? 0 : 2)] = part; other slot 0.
  // D = A x ones(4x16): D[m,n] = part(lane m) + part(lane m+16), replicated
  // over columns. Each lane sums its 8 D VGPRs (rows 0..7 or 8..15), then one
  // shfl_xor(16) merges the halves -> full wave sum, fp32 throughout.
  v2f a    = {part, 0.0f};
  v2f ones = {1.0f, 1.0f};
  v8f c    = {};
  c = __builtin_amdgcn_wmma_f32_16x16x4_f32(
      /*neg_a=*/false, a, /*neg_b=*/false, ones,
      /*c_mod=*/(short)0, c, /*reuse_a=*/false, /*reuse_b=*/false);
  float wsum = c[0] + c[1] + c[2] + c[3] + c[4] + c[5] + c[6] + c[7];
  wsum += __shfl_xor(wsum, 16);

  __shared__ float smem[8];
  int lane = threadIdx.x & 31;
  int wave = threadIdx.x >> 5;
  if (lane == 0) smem[wave] = wsum;
  __syncthreads();
  if (threadIdx.x == 0) {
    float tot = 0.0f;
    #pragma unroll
    for (int i = 0; i < 8; ++i) tot += smem[i];
    atomicAdd(&accum[b * NLVL + lvl], tot);
  }
}

// ---------------------------------------------------------------------------
// Kernel 3: finalize per-batch loss.
// ---------------------------------------------------------------------------
__global__ __launch_bounds__(64)
void la_final_kernel(const float* __restrict__ accum, const int* __restrict__ anyv,
                     float* __restrict__ out) {
  int b = threadIdx.x;
  if (b >= NB) return;
  const float inv[NLVL] = {1.0f/262144.0f, 1.0f/65536.0f, 1.0f/16384.0f,
                           1.0f/4096.0f,   1.0f/1024.0f};
  float s = 0.0f;
  #pragma unroll
  for (int l = 0; l < NLVL; ++l) s += accum[b * NLVL + l] * inv[l];
  out[b] = anyv[b] ? (s * 0.2f) : 0.0f;
}

// ---------------------------------------------------------------------------
extern "C" void kernel_launch(void* const* d_in, const int* in_sizes, int n_in,
                              void* d_out, int out_size, void* d_ws, size_t ws_size,
                              hipStream_t stream) {
  const float* att0  = (const float*)d_in[0];
  const float* att1  = (const float*)d_in[1];
  const float* att2  = (const float*)d_in[2];
  const float* att3  = (const float*)d_in[3];
  const float* att4  = (const float*)d_in[4];
  const float* bboxs = (const float*)d_in[5];
  const int*   hp    = (const int*)d_in[6];
  const int*   wp    = (const int*)d_in[7];
  float* out = (float*)d_out;

  char* ws = (char*)d_ws;
  float* accum = (float*)ws;                         // 320 floats  (1280 B)
  int4*  rect  = (int4*)(ws + 1280);                 // 320 int4    (5120 B)
  int*   flags = (int*)(ws + 1280 + 5120);           // 320 ints    (1280 B)
  int*   anyv  = (int*)(ws + 1280 + 5120 + 1280);    // 64 ints     ( 256 B)

  la_setup_kernel<<<1, 64, 0, stream>>>(bboxs, hp, wp, accum, rect, flags, anyv);
  la_bce_kernel<<<NB * BLOCKS_PER_BATCH, 256, 0, stream>>>(
      att0, att1, att2, att3, att4, accum, rect, flags);
  la_final_kernel<<<1, 64, 0, stream>>>(accum, anyv, out);
}